// VNTSimpleEncoder_56212531970676
// MI455X (gfx1250) — compile-verified
//
#include <hip/hip_runtime.h>
#include <hip/hip_bf16.h>

typedef __attribute__((ext_vector_type(2))) float v2f;
typedef __attribute__((ext_vector_type(8))) float v8f;

#define BB   4
#define NN   4096
#define CC   64
#define KNN  20
#define ROTC 116
#define EPSV 1e-6f

// ---------------- weight prep: row-normalize (VNT) + zero-pad rows to dstRows ----------------
__global__ void prep_norm_kernel(const float* __restrict__ src, float* __restrict__ dst,
                                 int srcRows, int dstRows, int cols) {
  int r = blockIdx.x * blockDim.x + threadIdx.x;
  if (r >= dstRows) return;
  if (r < srcRows) {
    float s = 0.f;
    for (int c = 0; c < cols; ++c) s += src[r * cols + c];
    float inv = 1.f / s;
    for (int c = 0; c < cols; ++c) dst[r * cols + c] = src[r * cols + c] * inv;
  } else {
    for (int c = 0; c < cols; ++c) dst[r * cols + c] = 0.f;
  }
}

// ---------------- weight prep: plain copy + zero-pad rows ----------------
__global__ void prep_copy_kernel(const float* __restrict__ src, float* __restrict__ dst,
                                 int srcRows, int dstRows, int cols) {
  int r = blockIdx.x * blockDim.x + threadIdx.x;
  if (r >= dstRows) return;
  if (r < srcRows) {
    for (int c = 0; c < cols; ++c) dst[r * cols + c] = src[r * cols + c];
  } else {
    for (int c = 0; c < cols; ++c) dst[r * cols + c] = 0.f;
  }
}

// ---------------- brute-force kNN, top-20 by neg squared distance (stable ties) ----------------
__global__ void __launch_bounds__(64) knn_kernel(const float* __restrict__ x,
                                                 int* __restrict__ idxOut) {
  __shared__ float px[NN], py[NN], pz[NN];
  __shared__ float tval[64 * KNN];
  __shared__ int   tidx[64 * KNN];
  const int blocksPerBatch = NN / 64;
  int b    = blockIdx.x / blocksPerBatch;
  int base = (blockIdx.x % blocksPerBatch) * 64;
  int t    = threadIdx.x;
  for (int j = t; j < NN; j += 64) {
    px[j] = x[(b * 3 + 0) * NN + j];
    py[j] = x[(b * 3 + 1) * NN + j];
    pz[j] = x[(b * 3 + 2) * NN + j];
  }
  __syncthreads();
  int   i  = base + t;
  float qx = px[i], qy = py[i], qz = pz[i];
  float sqi = qx * qx + qy * qy + qz * qz;
  float* v  = &tval[t * KNN];
  int*   id = &tidx[t * KNN];
  for (int k = 0; k < KNN; ++k) { v[k] = -3.4e38f; id[k] = 0; }
  float vmin = -3.4e38f;
  for (int j = 0; j < NN; ++j) {
    float bx = px[j], by = py[j], bz = pz[j];
    float d = 2.f * (qx * bx + qy * by + qz * bz) - sqi - (bx * bx + by * by + bz * bz);
    if (d > vmin) {
      int pos = KNN - 1;
      while (pos > 0 && v[pos - 1] < d) {
        v[pos] = v[pos - 1]; id[pos] = id[pos - 1]; --pos;
      }
      v[pos] = d; id[pos] = j;
      vmin = v[KNN - 1];
    }
  }
  for (int k = 0; k < KNN; ++k) idxOut[(b * NN + i) * KNN + k] = id[k];
}

// ---------------- edge features + Wpos VNT-lrelu + mean over k -> h0[B,64,3,N] ----------------
__global__ void __launch_bounds__(64) edge_kernel(const float* __restrict__ x,
                                                  const int* __restrict__ idx,
                                                  const float* __restrict__ WfN,
                                                  const float* __restrict__ Wd,
                                                  float* __restrict__ h0) {
  __shared__ float feat[KNN][3][3];  // [k][feat(nbr-ctr,ctr,cross)][vec]
  int b = blockIdx.x / NN;
  int n = blockIdx.x % NN;
  int t = threadIdx.x;
  float cx = x[(b * 3 + 0) * NN + n];
  float cy = x[(b * 3 + 1) * NN + n];
  float cz = x[(b * 3 + 2) * NN + n];
  if (t < KNN) {
    int   j  = idx[(b * NN + n) * KNN + t];
    float nx = x[(b * 3 + 0) * NN + j];
    float ny = x[(b * 3 + 1) * NN + j];
    float nz = x[(b * 3 + 2) * NN + j];
    feat[t][0][0] = nx - cx; feat[t][0][1] = ny - cy; feat[t][0][2] = nz - cz;
    feat[t][1][0] = cx;      feat[t][1][1] = cy;      feat[t][1][2] = cz;
    feat[t][2][0] = ny * cz - nz * cy;   // cross(nbr, ctr)
    feat[t][2][1] = nz * cx - nx * cz;
    feat[t][2][2] = nx * cy - ny * cx;
  }
  __syncthreads();
  int   c   = t;
  float wf0 = WfN[c * 3 + 0], wf1 = WfN[c * 3 + 1], wf2 = WfN[c * 3 + 2];
  float wd0 = Wd[c * 3 + 0],  wd1 = Wd[c * 3 + 1],  wd2 = Wd[c * 3 + 2];
  float a0 = 0.f, a1 = 0.f, a2 = 0.f;
  for (int k = 0; k < KNN; ++k) {
    float p0 = wf0 * feat[k][0][0] + wf1 * feat[k][1][0] + wf2 * feat[k][2][0];
    float p1 = wf0 * feat[k][0][1] + wf1 * feat[k][1][1] + wf2 * feat[k][2][1];
    float p2 = wf0 * feat[k][0][2] + wf1 * feat[k][1][2] + wf2 * feat[k][2][2];
    float d0 = wd0 * feat[k][0][0] + wd1 * feat[k][1][0] + wd2 * feat[k][2][0];
    float d1 = wd0 * feat[k][0][1] + wd1 * feat[k][1][1] + wd2 * feat[k][2][1];
    float d2 = wd0 * feat[k][0][2] + wd1 * feat[k][1][2] + wd2 * feat[k][2][2];
    float dot = p0 * d0 + p1 * d1 + p2 * d2;
    float dsq = d0 * d0 + d1 * d1 + d2 * d2;
    if (dot < 0.f) {
      float s = dot / (dsq + EPSV);
      p0 -= s * d0; p1 -= s * d1; p2 -= s * d2;
    }
    a0 += p0; a1 += p1; a2 += p2;
  }
  const float invK = 1.f / (float)KNN;
  h0[((b * CC + c) * 3 + 0) * NN + n] = a0 * invK;
  h0[((b * CC + c) * 3 + 1) * NN + n] = a1 * invK;
  h0[((b * CC + c) * 3 + 2) * NN + n] = a2 * invK;
}

// ---------------- WMMA channel GEMM + optional VN-lrelu / input-sub / mean-over-N ----------------
// Layout: in[B, CC, 3, NN].  One workgroup per (b, 16-point tile); wave rt handles out rows
// [rt*16, rt*16+16).  Weights must have >= ROWT*16 rows (zero-padded in prep).
// A frag (16x4 f32): lane<16 -> row=lane, k={k0,k0+1}; lane>=16 -> row=lane-16, k={k0+2,k0+3}
// B frag (4x16 f32): lane<16 -> col=lane, k={k0,k0+1}; lane>=16 -> col=lane-16, k={k0+2,k0+3}
// C/D (16x16 f32):   VGPR r, lane l -> (row = r + 8*(l/16), col = l%16)
template <bool LRELU, bool MEAN, bool SUBIN, int COUT>
__global__ void __launch_bounds__(((COUT + 15) / 16) * 32)
layer_kernel(const float* __restrict__ in, const float* __restrict__ inSub,
             const float* __restrict__ Wf, const float* __restrict__ Wd,
             float* __restrict__ out, int coffset) {
  constexpr int ROWT = (COUT + 15) / 16;
  int nt = blockIdx.x % (NN / 16);
  int b  = blockIdx.x / (NN / 16);
  int rt   = threadIdx.x >> 5;
  int lane = threadIdx.x & 31;
  int half = lane >> 4;
  int lh   = lane & 15;
  int n    = nt * 16 + lh;
  int arow = rt * 16 + lh;

  const float* pWf = Wf + arow * CC + 2 * half;
  const float* pWd = Wd + arow * CC + 2 * half;            // unused unless LRELU
  const float* pB  = in + (b * CC + 2 * half) * 3 * NN + n;
  const float* pS  = inSub + (b * CC + 2 * half) * 3 * NN + n;  // unused unless SUBIN

  v8f accP[3] = {v8f(0.f), v8f(0.f), v8f(0.f)};
  v8f accD[3] = {v8f(0.f), v8f(0.f), v8f(0.f)};

#pragma unroll
  for (int kk = 0; kk < CC / 4; ++kk) {
    v2f aF, aD;
    aF.x = pWf[kk * 4];
    aF.y = pWf[kk * 4 + 1];
    if (LRELU) {
      aD.x = pWd[kk * 4];
      aD.y = pWd[kk * 4 + 1];
    }
#pragma unroll
    for (int v = 0; v < 3; ++v) {
      const int o0 = kk * 4 * 3 * NN + v * NN;   // compile-time constant after unroll
      v2f bf;
      bf.x = pB[o0];
      bf.y = pB[o0 + 3 * NN];
      if (SUBIN) {
        bf.x -= pS[o0];
        bf.y -= pS[o0 + 3 * NN];
      }
      accP[v] = __builtin_amdgcn_wmma_f32_16x16x4_f32(false, aF, false, bf, (short)0,
                                                      accP[v], false, false);
      if (LRELU)
        accD[v] = __builtin_amdgcn_wmma_f32_16x16x4_f32(false, aD, false, bf, (short)0,
                                                        accD[v], false, false);
    }
  }

  const float invN = 1.f / (float)NN;
#pragma unroll
  for (int r = 0; r < 8; ++r) {
    int row = rt * 16 + r + 8 * half;
    float p0 = accP[0][r], p1 = accP[1][r], p2 = accP[2][r];
    if (LRELU) {
      float d0 = accD[0][r], d1 = accD[1][r], d2 = accD[2][r];
      float dot = p0 * d0 + p1 * d1 + p2 * d2;
      float dsq = d0 * d0 + d1 * d1 + d2 * d2;
      if (dot < 0.f) {
        float s = dot / (dsq + EPSV);
        p0 -= s * d0; p1 -= s * d1; p2 -= s * d2;
      }
    }
    if (!MEAN) {
      if (row < COUT) {
        out[((b * COUT + row) * 3 + 0) * NN + n] = p0;
        out[((b * COUT + row) * 3 + 1) * NN + n] = p1;
        out[((b * COUT + row) * 3 + 2) * NN + n] = p2;
      }
    } else {
      float s0 = p0, s1 = p1, s2 = p2;
#pragma unroll
      for (int m = 1; m < 16; m <<= 1) {
        s0 += __shfl_xor(s0, m, 32);
        s1 += __shfl_xor(s1, m, 32);
        s2 += __shfl_xor(s2, m, 32);
      }
      if (lh == 0 && row < COUT) {
        atomicAdd(&out[(b * 128 + coffset + row) * 3 + 0], s0 * invN);
        atomicAdd(&out[(b * 128 + coffset + row) * 3 + 1], s1 * invN);
        atomicAdd(&out[(b * 128 + coffset + row) * 3 + 2], s2 * invN);
      }
    }
  }
}

// ---------------- host orchestration ----------------
extern "C" void kernel_launch(void* const* d_in, const int* in_sizes, int n_in,
                              void* d_out, int out_size, void* d_ws, size_t ws_size,
                              hipStream_t stream) {
  (void)in_sizes; (void)n_in; (void)ws_size;
  const float* x      = (const float*)d_in[0];
  const float* Wpos_f = (const float*)d_in[1];
  const float* Wpos_d = (const float*)d_in[2];
  const float* Wc1_f  = (const float*)d_in[3];
  const float* Wc1_d  = (const float*)d_in[4];
  const float* Wc2_f  = (const float*)d_in[5];
  const float* Wc2_d  = (const float*)d_in[6];
  const float* Wt_f   = (const float*)d_in[7];
  const float* Wt_d   = (const float*)d_in[8];
  const float* W1_f   = (const float*)d_in[9];
  const float* W1_d   = (const float*)d_in[10];
  const float* W2_f   = (const float*)d_in[11];
  const float* W2_d   = (const float*)d_in[12];
  const float* W3     = (const float*)d_in[13];
  float* out = (float*)d_out;

  // workspace layout
  char* ws = (char*)d_ws;
  size_t off = 0;
  int* wsIdx   = (int*)(ws + off);   off += (size_t)BB * NN * KNN * sizeof(int);
  float* nWpos = (float*)(ws + off); off += 256 * sizeof(float);   // 64x3 normalized
  float* nWc1  = (float*)(ws + off); off += 4096 * sizeof(float);  // 64x64 normalized
  float* nWc2  = (float*)(ws + off); off += 4096 * sizeof(float);  // 64x64 normalized
  float* nWt   = (float*)(ws + off); off += 1024 * sizeof(float);  // 16x64 norm+pad
  float* pWtd  = (float*)(ws + off); off += 1024 * sizeof(float);  // 16x64 pad
  float* pW3   = (float*)(ws + off); off += 8192 * sizeof(float);  // 128x64 pad
  off = (off + 255) & ~(size_t)255;
  const size_t bufElems = (size_t)BB * CC * 3 * NN;
  float* buf0 = (float*)(ws + off);  off += bufElems * sizeof(float);
  float* buf1 = (float*)(ws + off);  off += bufElems * sizeof(float);
  float* buf2 = (float*)(ws + off);  off += bufElems * sizeof(float);

  // zero output accumulators (means are atomically accumulated into d_out)
  hipMemsetAsync(d_out, 0, (size_t)out_size * sizeof(float), stream);

  // weight prep (VNT normalization + zero-padding to row-tile multiples)
  prep_norm_kernel<<<1, 64, 0, stream>>>(Wpos_f, nWpos, 64, 64, 3);
  prep_norm_kernel<<<1, 64, 0, stream>>>(Wc1_f, nWc1, 64, 64, 64);
  prep_norm_kernel<<<1, 64, 0, stream>>>(Wc2_f, nWc2, 64, 64, 64);
  prep_norm_kernel<<<1, 64, 0, stream>>>(Wt_f, nWt, 12, 16, 64);
  prep_copy_kernel<<<1, 64, 0, stream>>>(Wt_d, pWtd, 12, 16, 64);
  prep_copy_kernel<<<2, 64, 0, stream>>>(W3, pW3, ROTC, 128, 64);

  // kNN indices
  knn_kernel<<<BB * (NN / 64), 64, 0, stream>>>(x, wsIdx);

  // edge features + pos_mlp + mean over k -> h0 (buf0)
  edge_kernel<<<BB * NN, 64, 0, stream>>>(x, wsIdx, nWpos, Wpos_d, buf0);

  const int grid = BB * (NN / 16);   // one workgroup per 16-point column tile
  // x2 = VNT-lrelu(h0, Wc1)                      buf0 -> buf1
  layer_kernel<true, false, false, 64><<<grid, 128, 0, stream>>>(buf0, buf0, nWc1, Wc1_d, buf1, 0);
  // xc = VNT-lrelu(x2, Wc2)                      buf1 -> buf2
  layer_kernel<true, false, false, 64><<<grid, 128, 0, stream>>>(buf1, buf1, nWc2, Wc2_d, buf2, 0);
  // trans_latent = mean_N( VNT-lrelu(xc, Wt) )   buf2 -> d_out[:,116:128,:]
  layer_kernel<true, true, false, 12><<<grid, 32, 0, stream>>>(buf2, buf2, nWt, pWtd, out, 116);
  // h = lrelu(h0 - xc, W1)   (sub fused)         (buf0 - buf2) -> buf1
  layer_kernel<true, false, true, 64><<<grid, 128, 0, stream>>>(buf0, buf2, W1_f, W1_d, buf1, 0);
  // h = lrelu(h, W2)                             buf1 -> buf0
  layer_kernel<true, false, false, 64><<<grid, 128, 0, stream>>>(buf1, buf1, W2_f, W2_d, buf0, 0);
  // rot_latent = mean_N( W3 h )                  buf0 -> d_out[:,0:116,:]
  layer_kernel<false, true, false, ROTC><<<grid, 256, 0, stream>>>(buf0, buf0, pW3, pW3, out, 0);
}